// Actor2LS_79001628443219
// MI455X (gfx1250) — compile-verified
//
#include <hip/hip_runtime.h>

// Actor2LS fused pipeline for gfx1250 (MI455X), fp32 WMMA (V_WMMA_F32_16X16X4_F32).
// - Exact mask sparsity (dist<=6 over 100x100 uniform centers => ~1% pairs active)
// - Deterministic per-row compaction (no float atomics)
// - 2 M-tiles per block in row GEMMs => B fragments reused across two wmma chains
// - float4 (b128) staging loads, parallel tile GroupNorm reduction

#define D      128
#define NLSC   800
#define NAC    48
#define EPSGN  1e-5f
#define DIST2  36.0f

typedef float v2f __attribute__((ext_vector_type(2)));
typedef float v4f __attribute__((ext_vector_type(4)));
typedef float v8f __attribute__((ext_vector_type(8)));

// ---- WMMA fp32 16x16x4. Layouts per CDNA5 ISA 7.12.2 ----
// A 16x4 f32: lanes 0-15: M=lane, VGPR0=K0,VGPR1=K1; lanes 16-31: M=lane-16, K2/K3.
// B 4x16 f32: mirrored over N.  C/D: VGPR r holds M = r + 8*(lane>=16), N = lane&15.

__device__ __forceinline__ v8f wmma4(v2f a, v2f b, v8f c) {
  return __builtin_amdgcn_wmma_f32_16x16x4_f32(false, a, false, b, (short)0, c,
                                               false, false);
}

template<int K, int LDA>
__device__ __forceinline__ v8f wmma_tile(const float* sA, const float* __restrict__ B,
                                         int nbase, v8f acc) {
  const int lane = threadIdx.x & 31;
  const int half = lane >> 4;
  const int l    = lane & 15;
#pragma unroll
  for (int k0 = 0; k0 < K; k0 += 4) {
    const int ka = k0 + 2 * half;
    v2f a, b;
    a.x = sA[l * LDA + ka];
    a.y = sA[l * LDA + ka + 1];
    b.x = B[ka * D + nbase + l];
    b.y = B[(ka + 1) * D + nbase + l];
    acc = wmma4(a, b, acc);
  }
  return acc;
}

__device__ __forceinline__ void tile_to_lds(float* sOut, int ldo, int nbase, v8f acc) {
  const int lane = threadIdx.x & 31;
  const int half = lane >> 4;
  const int l    = lane & 15;
#pragma unroll
  for (int r = 0; r < 8; ++r)
    sOut[(r + 8 * half) * ldo + nbase + l] = acc[r];
}

// ---- Row GEMM: OUT[m0:m0+32, :] = A[m0:m0+32, :] @ B[128x128] (+C0) ----
// 2 M-tiles per block: B fragment loaded once, used by two wmma chains.
__global__ __launch_bounds__(256) void gemm_x_kernel(const float* __restrict__ A,
                                                     const float* __restrict__ B,
                                                     const float* __restrict__ C0,
                                                     float* __restrict__ OUT) {
  __shared__ v4f sA4[32 * 32];
  float* sA = (float*)sA4;
  const int m0 = blockIdx.x * 32;
  const int t  = threadIdx.x;
  const v4f* Av = (const v4f*)(A + (size_t)m0 * D);   // flat contiguous 32x128 tile
#pragma unroll
  for (int k = 0; k < 4; ++k) sA4[t + k * 256] = Av[t + k * 256];
  __syncthreads();

  const int wave  = t >> 5;
  const int nbase = wave * 16;
  const int lane  = t & 31;
  const int half  = lane >> 4;
  const int l     = lane & 15;

  v8f acc0 = {}, acc1 = {};
  if (C0) {
#pragma unroll
    for (int r = 0; r < 8; ++r) {
      acc0[r] = C0[(m0 + r + 8 * half) * D + nbase + l];
      acc1[r] = C0[(m0 + 16 + r + 8 * half) * D + nbase + l];
    }
  }
#pragma unroll
  for (int k0 = 0; k0 < D; k0 += 4) {
    const int ka = k0 + 2 * half;
    v2f bf, a0, a1;
    bf.x = B[ka * D + nbase + l];
    bf.y = B[(ka + 1) * D + nbase + l];
    a0.x = sA[l * D + ka];
    a0.y = sA[l * D + ka + 1];
    a1.x = sA[(16 + l) * D + ka];
    a1.y = sA[(16 + l) * D + ka + 1];
    acc0 = wmma4(a0, bf, acc0);
    acc1 = wmma4(a1, bf, acc1);
  }
#pragma unroll
  for (int r = 0; r < 8; ++r) {
    OUT[(m0 + r + 8 * half) * D + nbase + l]      = acc0[r];
    OUT[(m0 + 16 + r + 8 * half) * D + nbase + l] = acc1[r];
  }
}

// ---- Meta fuse GEMM: OUT = cat(feat, turn, control, intersect) @ meta_w (132x128) ----
__global__ __launch_bounds__(256) void gemm_meta_kernel(const float* __restrict__ feat,
                                                        const float* __restrict__ turn,
                                                        const float* __restrict__ control,
                                                        const float* __restrict__ intersect,
                                                        const float* __restrict__ metaW,
                                                        float* __restrict__ OUT) {
  __shared__ float sA[16 * 132];
  const int m0 = blockIdx.x * 16;
  const int t  = threadIdx.x;
  for (int e = t; e < 16 * 132; e += 256) {
    const int m   = e / 132, c = e % 132;
    const int row = m0 + m;
    float v;
    if (c < 128)       v = feat[row * D + c];
    else if (c < 130)  v = turn[row * 2 + (c - 128)];
    else if (c == 130) v = control[row];
    else               v = intersect[row];
    sA[m * 132 + c] = v;
  }
  __syncthreads();
  const int wave  = t >> 5;
  const int nbase = wave * 16;
  const int lane  = t & 31;
  const int half  = lane >> 4;
  const int l     = lane & 15;
  v8f acc = {};
  acc = wmma_tile<132, 132>(sA, metaW, nbase, acc);
#pragma unroll
  for (int r = 0; r < 8; ++r)
    OUT[(m0 + r + 8 * half) * D + nbase + l] = acc[r];
}

// ---- Per-row GroupNorm(1 group over C=128) + optional residual + ReLU ----
__global__ __launch_bounds__(128) void gn_rows_kernel(const float* __restrict__ in,
                                                      const float* __restrict__ g,
                                                      const float* __restrict__ bt,
                                                      const float* __restrict__ res,
                                                      float* __restrict__ out) {
  __shared__ float red[D];
  const int row = blockIdx.x;
  const int t   = threadIdx.x;
  const float v = in[row * D + t];
  red[t] = v;
  __syncthreads();
  for (int s = 64; s > 0; s >>= 1) {
    if (t < s) red[t] += red[t + s];
    __syncthreads();
  }
  const float mean = red[0] * (1.0f / D);
  __syncthreads();
  const float dv = v - mean;
  red[t] = dv * dv;
  __syncthreads();
  for (int s = 64; s > 0; s >>= 1) {
    if (t < s) red[t] += red[t + s];
    __syncthreads();
  }
  const float var = red[0] * (1.0f / D);
  float y = dv * rsqrtf(var + EPSGN) * g[t] + bt[t];
  if (res) y += res[row * D + t];
  out[row * D + t] = fmaxf(y, 0.0f);
}

// ---- GN(+ReLU) over a 16x128 LDS tile, parallel reduction (16 threads/row) ----
__device__ __forceinline__ void gn_tile(const float* sIn, const float* __restrict__ g,
                                        const float* __restrict__ bt, float* sOut, int ldo,
                                        float* sR1, float* sR2, float* rMean, float* rStd) {
  const int t = threadIdx.x;
  const int r = t >> 4, j = t & 15;
  float s1 = 0.f, s2 = 0.f;
#pragma unroll
  for (int k = 0; k < 8; ++k) {
    const float v = sIn[r * D + j * 8 + k];
    s1 += v;
    s2 += v * v;
  }
  sR1[t] = s1; sR2[t] = s2;
  __syncthreads();
  if (j < 8) { sR1[t] += sR1[t + 8]; sR2[t] += sR2[t + 8]; }
  __syncthreads();
  if (j < 4) { sR1[t] += sR1[t + 4]; sR2[t] += sR2[t + 4]; }
  __syncthreads();
  if (j < 2) { sR1[t] += sR1[t + 2]; sR2[t] += sR2[t + 2]; }
  __syncthreads();
  if (j == 0) {
    const float mean = (sR1[t] + sR1[t + 1]) * (1.0f / D);
    const float ex2  = (sR2[t] + sR2[t + 1]) * (1.0f / D);
    rMean[r] = mean;
    rStd[r]  = rsqrtf(fmaxf(ex2 - mean * mean, 0.0f) + EPSGN);
  }
  __syncthreads();
  for (int e = t; e < 16 * D; e += 256) {
    const int m = e >> 7, c = e & 127;
    const float y = (sIn[e] - rMean[m]) * rStd[m] * g[c] + bt[c];
    sOut[m * ldo + c] = fmaxf(y, 0.0f);
  }
  __syncthreads();
}

// ---- Pair MLP + masked scatter-sum: one block per (b, ls) row ----
__global__ __launch_bounds__(256) void pair_msg_kernel(
    const float* __restrict__ ls_ctrs, const float* __restrict__ actor_ctrs,
    const float* __restrict__ q, const float* __restrict__ actors,
    const float* __restrict__ dw0, const float* __restrict__ db0,
    const float* __restrict__ dw1,
    const float* __restrict__ dgnw, const float* __restrict__ dgnb,
    const float* __restrict__ cw0,
    const float* __restrict__ cgnw, const float* __restrict__ cgnb,
    const float* __restrict__ cw1, float* __restrict__ msg) {
  __shared__ v4f sA4[16 * 96];     // concat [gn(d@dw1) | q | actors], row stride 384
  __shared__ v4f sD4[16 * 32];
  __shared__ float sE[16 * D];
  __shared__ float sR1[256], sR2[256];
  __shared__ float sMsg[D];
  __shared__ float aDx[NAC], aDy[NAC];
  __shared__ int   aIdx[NAC];
  __shared__ int   sCnt;
  __shared__ float rMean[16], rStd[16];

  float* sA = (float*)sA4;
  float* sD = (float*)sD4;

  const int row = blockIdx.x;
  const int b   = row / NLSC;
  const int t   = threadIdx.x;

  if (t == 0) {
    const float lx = ls_ctrs[row * 2], ly = ls_ctrs[row * 2 + 1];
    int cnt = 0;
    for (int a = 0; a < NAC; ++a) {               // ascending order => deterministic sum
      const float dx = lx - actor_ctrs[(b * NAC + a) * 2];
      const float dy = ly - actor_ctrs[(b * NAC + a) * 2 + 1];
      if (dx * dx + dy * dy <= DIST2) {
        aIdx[cnt] = a; aDx[cnt] = dx; aDy[cnt] = dy; ++cnt;
      }
    }
    sCnt = cnt;
  }
  if (t < D) sMsg[t] = 0.0f;
  __syncthreads();
  const int count = sCnt;

  if (count > 0) {                                // uniform branch: EXEC full for WMMA
    const int wave  = t >> 5;
    const int nbase = wave * 16;
    const v4f* dw04 = (const v4f*)dw0;
    const v4f* db04 = (const v4f*)db0;
    const v4f* q4   = (const v4f*)q;
    const v4f* act4 = (const v4f*)actors;

    for (int base = 0; base < count; base += 16) {
      const int P = (count - base < 16) ? (count - base) : 16;
      // stage 1: d = relu(dvec @ dw0 + db0)   (K=2, VALU, float4)
      for (int e4 = t; e4 < 16 * 32; e4 += 256) {
        const int m = e4 >> 5, j = e4 & 31;
        v4f v = {};
        if (m < P) {
          const v4f w0 = dw04[j];
          const v4f w1 = dw04[32 + j];
          const v4f bb = db04[j];
          const float dx = aDx[base + m], dy = aDy[base + m];
          v.x = fmaxf(dx * w0.x + dy * w1.x + bb.x, 0.0f);
          v.y = fmaxf(dx * w0.y + dy * w1.y + bb.y, 0.0f);
          v.z = fmaxf(dx * w0.z + dy * w1.z + bb.z, 0.0f);
          v.w = fmaxf(dx * w0.w + dy * w1.w + bb.w, 0.0f);
        }
        sD4[e4] = v;
      }
      __syncthreads();
      // stage 2: E = d @ dw1 ; relu(gn(E)) -> sA[:, 0:128]
      {
        v8f acc = {};
        acc = wmma_tile<D, D>(sD, dw1, nbase, acc);
        tile_to_lds(sE, D, nbase, acc);
      }
      __syncthreads();
      gn_tile(sE, dgnw, dgnb, sA, 384, sR1, sR2, rMean, rStd);
      // fill q (same row for all pairs of this ls) and gathered actor rows
      if (t < 32) {
        const v4f qv = q4[row * 32 + t];
#pragma unroll
        for (int m = 0; m < 16; ++m) sA4[m * 96 + 32 + t] = qv;
      }
      for (int e4 = t; e4 < 16 * 32; e4 += 256) {
        const int m = e4 >> 5, j = e4 & 31;
        v4f av = {};
        if (m < P) av = act4[(b * NAC + aIdx[base + m]) * 32 + j];
        sA4[m * 96 + 64 + j] = av;
      }
      __syncthreads();
      // stage 3: F = c @ cw0 (K=384) ; relu(gn(F)) -> sD
      {
        v8f acc = {};
        acc = wmma_tile<384, 384>(sA, cw0, nbase, acc);
        tile_to_lds(sE, D, nbase, acc);
      }
      __syncthreads();
      gn_tile(sE, cgnw, cgnb, sD, D, sR1, sR2, rMean, rStd);
      // stage 4: H = f @ cw1 -> sE
      {
        v8f acc = {};
        acc = wmma_tile<D, D>(sD, cw1, nbase, acc);
        tile_to_lds(sE, D, nbase, acc);
      }
      __syncthreads();
      // ordered accumulation of active rows (deterministic)
      if (t < D) {
        float s = sMsg[t];
        for (int m = 0; m < P; ++m) s += sE[m * D + t];
        sMsg[t] = s;
      }
      __syncthreads();
    }
  }
  if (t < D) msg[row * D + t] = sMsg[t];
}

extern "C" void kernel_launch(void* const* d_in, const int* in_sizes, int n_in,
                              void* d_out, int out_size, void* d_ws, size_t ws_size,
                              hipStream_t stream) {
  const float* feat       = (const float*)d_in[0];
  const float* turn       = (const float*)d_in[1];
  const float* control    = (const float*)d_in[2];
  const float* intersect  = (const float*)d_in[3];
  const float* ls_ctrs    = (const float*)d_in[4];
  const float* actors     = (const float*)d_in[5];
  const float* actor_ctrs = (const float*)d_in[6];
  const float* meta_w     = (const float*)d_in[7];
  const float* mgn_w      = (const float*)d_in[8];
  const float* mgn_b      = (const float*)d_in[9];
  const float* dw0        = (const float*)d_in[10];
  const float* db0        = (const float*)d_in[11];
  const float* dw1        = (const float*)d_in[12];
  const float* dgn_w      = (const float*)d_in[13];
  const float* dgn_b      = (const float*)d_in[14];
  const float* qw         = (const float*)d_in[15];
  const float* qgn_w      = (const float*)d_in[16];
  const float* qgn_b      = (const float*)d_in[17];
  const float* cw0        = (const float*)d_in[18];
  const float* cgn_w      = (const float*)d_in[19];
  const float* cgn_b      = (const float*)d_in[20];
  const float* cw1        = (const float*)d_in[21];
  const float* aw         = (const float*)d_in[22];
  const float* ngn_w      = (const float*)d_in[23];
  const float* ngn_b      = (const float*)d_in[24];
  const float* lw         = (const float*)d_in[25];
  const float* lgn_w      = (const float*)d_in[26];
  const float* lgn_b      = (const float*)d_in[27];

  const int M = in_sizes[0] / D;           // 6400 = B*NLS
  const size_t ROWS = (size_t)M * D;

  float* X   = (float*)d_ws;               // current features / residual
  float* Y   = X + ROWS;
  float* TMP = Y + ROWS;
  float* Q   = TMP + ROWS;
  float* MSG = Q + ROWS;

  const dim3 gGemmX(M / 32), bGemm(256);
  const dim3 gMeta(M / 16);
  const dim3 gRows(M), bRows(D);
  const dim3 gPair(M), bPair(256);

  // meta fuse: x = relu(gn(cat(feat, meta) @ meta_w))
  gemm_meta_kernel<<<gMeta, bGemm, 0, stream>>>(feat, turn, control, intersect, meta_w, TMP);
  gn_rows_kernel<<<gRows, bRows, 0, stream>>>(TMP, mgn_w, mgn_b, nullptr, X);

  for (int i = 0; i < 2; ++i) {
    const float* dw0_i = dw0 + i * 2 * D;
    const float* db0_i = db0 + i * D;
    const float* dw1_i = dw1 + i * D * D;
    const float* qw_i  = qw + i * D * D;
    const float* cw0_i = cw0 + i * 3 * D * D;
    const float* cw1_i = cw1 + i * D * D;
    const float* aw_i  = aw + i * D * D;
    const float* lw_i  = lw + i * D * D;

    // q = relu(gn(x @ qw))
    gemm_x_kernel<<<gGemmX, bGemm, 0, stream>>>(X, qw_i, nullptr, TMP);
    gn_rows_kernel<<<gRows, bRows, 0, stream>>>(TMP, qgn_w + i * D, qgn_b + i * D, nullptr, Q);
    // msg = masked scatter-sum of pair MLP
    pair_msg_kernel<<<gPair, bPair, 0, stream>>>(ls_ctrs, actor_ctrs, Q, actors,
                                                 dw0_i, db0_i, dw1_i,
                                                 dgn_w + i * D, dgn_b + i * D,
                                                 cw0_i, cgn_w + i * D, cgn_b + i * D,
                                                 cw1_i, MSG);
    // x' = relu(gn(x @ aw + msg))
    gemm_x_kernel<<<gGemmX, bGemm, 0, stream>>>(X, aw_i, MSG, TMP);
    gn_rows_kernel<<<gRows, bRows, 0, stream>>>(TMP, ngn_w + i * D, ngn_b + i * D, nullptr, Y);
    // x = relu(gn(x' @ lw) + res)
    gemm_x_kernel<<<gGemmX, bGemm, 0, stream>>>(Y, lw_i, nullptr, TMP);
    float* outp = (i == 1) ? (float*)d_out : X;
    gn_rows_kernel<<<gRows, bRows, 0, stream>>>(TMP, lgn_w + i * D, lgn_b + i * D, X, outp);
  }
}